// OHEM_BCELoss_9895604649992
// MI455X (gfx1250) — compile-verified
//
#include <hip/hip_runtime.h>
#include <stdint.h>

// OHEM BCE loss for MI455X (gfx1250).
// Pass 1: async-DMA (global_load_async_to_lds_b128) double-buffered streaming,
//         v_log_f32 BCE, per-WG LDS histogram (count + fixed-point sum).
// Pass 2: cross-WG histogram reduce + suffix scan -> top-num_neg selection.
// All reductions are integer fixed-point (x 2^24) => deterministic.

typedef unsigned int u32;
typedef unsigned long long u64;

#define NB     1024            // histogram bins over loss range [0, MAXL)
#define BLOCK  256             // threads per workgroup (8 waves of 32)
#define GRID1  512             // workgroups for pass 1
#define CHUNK  (BLOCK * 4)     // elements per chunk (float4 per thread)
#define MAXL   16.7f           // max possible bce: -log(2^-24) = 16.64
#define SCALE_F 16777216.0f    // 2^24 fixed-point scale

// ---- CDNA5 async global -> LDS copy (ASYNCcnt-tracked DMA path) ------------
__device__ __forceinline__ void async_copy_b128(u32 lds_off, const void* gptr) {
    asm volatile("global_load_async_to_lds_b128 %0, %1, off"
                 :: "v"(lds_off), "v"((unsigned long long)(uintptr_t)gptr)
                 : "memory");
}
__device__ __forceinline__ void wait_async_le2() {
    asm volatile("s_wait_asynccnt 2" ::: "memory");
}
__device__ __forceinline__ void wait_async_0() {
    asm volatile("s_wait_asynccnt 0" ::: "memory");
}
__device__ __forceinline__ void wait_ds_0() {
    asm volatile("s_wait_dscnt 0" ::: "memory");
}

__global__ __launch_bounds__(BLOCK) void ohem_pass1(
    const float* __restrict__ in, const float* __restrict__ tg, int N,
    u64* __restrict__ wsum,      // [GRID1][NB] fixed-point bce sums (negatives)
    u32* __restrict__ wcnt,      // [GRID1][NB] counts (negatives)
    u64* __restrict__ wpossum,   // [GRID1]
    u32* __restrict__ wposcnt)   // [GRID1]
{
    __shared__ alignas(16) float s_in[2][CHUNK];
    __shared__ alignas(16) float s_tg[2][CHUNK];
    __shared__ u32 s_cnt[NB];
    __shared__ u64 s_sum[NB];
    __shared__ u32 s_pc;
    __shared__ u64 s_ps;

    const int tid = threadIdx.x;
    const int bid = blockIdx.x;
    const int nWG = gridDim.x;
    const int numChunks = N / CHUNK;

    const u32 ldsInBase = (u32)(uintptr_t)(&s_in[0][0]);
    const u32 ldsTgBase = (u32)(uintptr_t)(&s_tg[0][0]);
    const u32 myLds16   = (u32)tid * 16u;

    // Prefetch first chunk ASAP (overlaps with histogram zeroing below).
    int c = bid;
    if (c < numChunks) {
        size_t eb = (size_t)c * CHUNK + (size_t)tid * 4;
        async_copy_b128(ldsInBase + myLds16, in + eb);
        async_copy_b128(ldsTgBase + myLds16, tg + eb);
    }

    for (int i = tid; i < NB; i += BLOCK) { s_cnt[i] = 0u; s_sum[i] = 0ull; }
    if (tid == 0) { s_pc = 0u; s_ps = 0ull; }
    __syncthreads();

    u32 myPC = 0; u64 myPS = 0;

    auto procElem = [&](float x, float t) {
        float p   = fminf(fmaxf(x, 1e-7f), 1.0f - 1e-7f);
        bool pos  = (t > 0.5f);
        float v   = pos ? p : (1.0f - p);       // v in (0,1], strictly < 1 effectively
        float bce = -__logf(v);                  // v_log_f32 path; bce >= 0
        u32 fx    = (u32)(bce * SCALE_F);        // <= 16.64*2^24 < 2^32
        if (pos) {
            myPC += 1u; myPS += (u64)fx;
        } else {
            int bin = (int)(bce * ((float)NB / MAXL));
            if (bin > NB - 1) bin = NB - 1;
            atomicAdd(&s_cnt[bin], 1u);
            atomicAdd(&s_sum[bin], (u64)fx);
        }
    };

    int buf = 0;
    for (; c < numChunks; c += nWG) {
        int nc = c + nWG;
        if (nc < numChunks) {
            // WAR fence: previous reads of buf^1 must finish before DMA overwrites it.
            wait_ds_0();
            int nb = buf ^ 1;
            u32 bOff = (u32)(nb * CHUNK * 4);
            size_t eb = (size_t)nc * CHUNK + (size_t)tid * 4;
            async_copy_b128(ldsInBase + bOff + myLds16, in + eb);
            async_copy_b128(ldsTgBase + bOff + myLds16, tg + eb);
            wait_async_le2();   // in-order completion => current chunk's 2 ops done
        } else {
            wait_async_0();
        }
        float4 x = ((const float4*)(&s_in[buf][0]))[tid];  // same bytes this lane DMA'd
        float4 t = ((const float4*)(&s_tg[buf][0]))[tid];
        procElem(x.x, t.x); procElem(x.y, t.y);
        procElem(x.z, t.z); procElem(x.w, t.w);
        buf ^= 1;
    }

    // Scalar tail (N not multiple of CHUNK).
    for (long long i = (long long)numChunks * CHUNK + (long long)bid * BLOCK + tid;
         i < (long long)N; i += (long long)nWG * BLOCK) {
        procElem(in[i], tg[i]);
    }

    __syncthreads();
    if (myPC) { atomicAdd(&s_pc, myPC); atomicAdd(&s_ps, myPS); }
    __syncthreads();

    // Plain (deterministic) per-WG flush; pass 2 reduces across WGs.
    for (int i = tid; i < NB; i += BLOCK) {
        wsum[(size_t)bid * NB + i] = s_sum[i];
        wcnt[(size_t)bid * NB + i] = s_cnt[i];
    }
    if (tid == 0) { wpossum[bid] = s_ps; wposcnt[bid] = s_pc; }
}

__global__ __launch_bounds__(BLOCK) void ohem_pass2(
    const u64* __restrict__ wsum, const u32* __restrict__ wcnt,
    const u64* __restrict__ wpossum, const u32* __restrict__ wposcnt,
    float* __restrict__ out, int N, int grid1)
{
    __shared__ u64 totSum[NB];
    __shared__ u32 totCnt[NB];
    __shared__ u64 ssum[BLOCK + 1];
    __shared__ u32 scnt[BLOCK + 1];
    __shared__ u32 sPC;
    __shared__ u64 sPS;

    const int tid = threadIdx.x;
    if (tid == 0) { sPC = 0u; sPS = 0ull; scnt[BLOCK] = 0u; ssum[BLOCK] = 0ull; }
    __syncthreads();

    // Reduce per-WG positive stats (integer => order-independent).
    u32 pc = 0; u64 ps = 0;
    for (int w = tid; w < grid1; w += BLOCK) { pc += wposcnt[w]; ps += wpossum[w]; }
    if (pc | (u32)(ps != 0)) { atomicAdd(&sPC, pc); atomicAdd(&sPS, ps); }

    // Reduce histograms: bins strided by tid => lane-consecutive (coalesced).
    for (int b = tid; b < NB; b += BLOCK) {
        u64 s = 0; u32 cc = 0;
        for (int w = 0; w < grid1; ++w) {
            s  += wsum[(size_t)w * NB + b];
            cc += wcnt[(size_t)w * NB + b];
        }
        totSum[b] = s; totCnt[b] = cc;
    }
    __syncthreads();

    // Each thread owns BPT contiguous bins (ordered by loss value).
    const int BPT = NB / BLOCK;   // 4
    u32 cArr[NB / BLOCK]; u64 sArr[NB / BLOCK];
    u32 myC = 0; u64 myS = 0;
    for (int k = 0; k < BPT; ++k) {
        int bin = tid * BPT + k;
        cArr[k] = totCnt[bin]; sArr[k] = totSum[bin];
        myC += cArr[k]; myS += sArr[k];
    }
    scnt[tid] = myC; ssum[tid] = myS;
    __syncthreads();

    // Inclusive suffix scan over thread-chunks (Hillis-Steele).
    for (int off = 1; off < BLOCK; off <<= 1) {
        u32 ac = 0; u64 as = 0;
        if (tid + off < BLOCK) { ac = scnt[tid + off]; as = ssum[tid + off]; }
        __syncthreads();
        scnt[tid] += ac; ssum[tid] += as;
        __syncthreads();
    }

    const u32 numPos   = sPC;
    const u64 posSum   = sPS;
    const u32 totalNeg = scnt[0];

    // Reference: num_neg = min(int32(3.0f * num_pos), total - num_pos)
    long long nn    = (long long)(3.0f * (float)numPos);
    long long avail = (long long)N - (long long)numPos;
    long long num_neg = (nn < avail) ? nn : avail;

    if (num_neg <= 0) {                     // no negatives kept
        if (tid == 0) {
            double denom = (double)((numPos > 0u) ? numPos : 1u);
            out[0] = (float)(((double)posSum / 16777216.0) / denom);
        }
        return;
    }
    if ((u64)num_neg >= (u64)totalNeg) {    // keep all negatives (expected here)
        if (tid == 0) {
            double denom = (double)((u64)numPos + (u64)num_neg);
            out[0] = (float)((((double)posSum + (double)ssum[0]) / 16777216.0) / denom);
        }
        return;
    }

    // Threshold chunk: above < num_neg <= inclusive-suffix  (unique thread).
    const u32 nneg  = (u32)num_neg;
    const u32 above = scnt[tid + 1];
    if (above < nneg && scnt[tid] >= nneg) {
        u32 running = above; u64 runSum = ssum[tid + 1];
        for (int k = BPT - 1; k >= 0; --k) {       // scan bins hardest-first
            u32 c0 = cArr[k];
            if (running + c0 >= nneg) {
                u32 part = nneg - running;          // 1..c0 elements of this bin
                double nsum = (double)runSum +
                              (double)part * ((double)sArr[k] / (double)c0);
                double denom = (double)((u64)numPos + (u64)nneg);
                out[0] = (float)((((double)posSum + nsum) / 16777216.0) / denom);
                break;
            }
            running += c0; runSum += sArr[k];
        }
    }
}

extern "C" void kernel_launch(void* const* d_in, const int* in_sizes, int n_in,
                              void* d_out, int out_size, void* d_ws, size_t ws_size,
                              hipStream_t stream) {
    const float* in = (const float*)d_in[0];
    const float* tg = (const float*)d_in[1];
    const int N = in_sizes[0];

    char* ws = (char*)d_ws;
    u64* wsum    = (u64*)(ws);                                          // GRID1*NB*8
    u32* wcnt    = (u32*)(ws + (size_t)GRID1 * NB * 8);                 // GRID1*NB*4
    u64* wpossum = (u64*)(ws + (size_t)GRID1 * NB * 12);                // GRID1*8
    u32* wposcnt = (u32*)(ws + (size_t)GRID1 * NB * 12 + GRID1 * 8);    // GRID1*4
    // total workspace used: 512*1024*12 + 512*12 ~= 6.3 MB

    ohem_pass1<<<GRID1, BLOCK, 0, stream>>>(in, tg, N, wsum, wcnt, wpossum, wposcnt);
    ohem_pass2<<<1, BLOCK, 0, stream>>>(wsum, wcnt, wpossum, wposcnt,
                                        (float*)d_out, N, GRID1);
}